// GraphConvolution_35304631173973
// MI455X (gfx1250) — compile-verified
//
#include <hip/hip_runtime.h>
#include <cstdint>
#include <cstddef>

typedef __attribute__((ext_vector_type(2))) float v2f;
typedef __attribute__((ext_vector_type(4))) float v4f;
typedef __attribute__((ext_vector_type(8))) float v8f;
typedef int v4i __attribute__((__vector_size__(16)));

constexpr int kN = 16384;   // nodes
constexpr int kD = 128;     // D_IN == D_OUT

constexpr int kBlockM = 64; // output rows per block (GEMM2)
constexpr int kKC = 32;     // K chunk staged in LDS
constexpr int kLda = 36;    // padded LDS stride (floats) for A slab: 16B-aligned, conflict-free
constexpr int kLds = 136;   // padded LDS stride (floats) for S slab: 16B-aligned, conflict-free
constexpr int kChunks = kN / kKC;

#if defined(__gfx1250__) && \
    __has_builtin(__builtin_amdgcn_global_load_async_to_lds_b128) && \
    __has_builtin(__builtin_amdgcn_s_wait_asynccnt)
#define HAVE_ASYNC_LDS 1
#else
#define HAVE_ASYNC_LDS 0
#endif

// ---------------------------------------------------------------------------
// GEMM1: support = X[16384,128] @ W[128,128]   (one wave per 16x16 tile)
// ---------------------------------------------------------------------------
__global__ __launch_bounds__(128, 1) void gcn_support(const float* __restrict__ X,
                                                      const float* __restrict__ W,
                                                      float* __restrict__ S) {
  const int wave = threadIdx.x >> 5;
  const int lane = threadIdx.x & 31;
  const int lm   = lane & 15;
  const int half = lane >> 4;

  const int tile = blockIdx.x * 4 + wave;   // kD/16 = 8 tiles per row strip
  const int m0 = (tile >> 3) * 16;
  const int n0 = (tile & 7) * 16;

  v8f acc = v8f{};
  // A frag: lane(lm) = row M, half selects K pair {0,1} vs {2,3}
  const float* xp = X + (size_t)(m0 + lm) * kD + 2 * half;
  // B frag: lane(lm) = col N, half selects K row +0/+1 vs +2/+3
  const float* wp = W + (size_t)(2 * half) * kD + n0 + lm;

#pragma unroll 4
  for (int k = 0; k < kD; k += 4) {
    v2f a = *(const v2f*)(xp + k);          // X[m][k+2h], X[m][k+2h+1]
    v2f b;
    b.x = wp[(size_t)k * kD];               // W[k+2h][n]
    b.y = wp[(size_t)(k + 1) * kD];         // W[k+2h+1][n]
    acc = __builtin_amdgcn_wmma_f32_16x16x4_f32(false, a, false, b,
                                                (short)0, acc, false, false);
  }

  // D layout: VGPR i -> row m0 + i + 8*half, col n0 + lm
  float* op = S + (size_t)(m0 + 8 * half) * kD + n0 + lm;
#pragma unroll
  for (int i = 0; i < 8; ++i) op[(size_t)i * kD] = acc[i];
}

// ---------------------------------------------------------------------------
// GEMM2: out = A[16384,16384] @ support[16384,128]
// Block: 128 threads (4 waves), 64 rows x 128 cols of output.
// A streamed once from HBM, support slab staged via LDS.
// Async path: double-buffered GLOBAL_LOAD_ASYNC_TO_LDS_B128 (ASYNCcnt).
// ---------------------------------------------------------------------------
#if HAVE_ASYNC_LDS

#define ASYNC_GPTR(p) ((__attribute__((address_space(1))) v4i*)(p))
#define ASYNC_LPTR(p) ((__attribute__((address_space(3))) v4i*)(p))

__device__ __forceinline__ void stage_async(const float* __restrict__ A,
                                            const float* __restrict__ S,
                                            int mBase, int k0, int tid,
                                            float* bA, float* bS) {
  // A slab: 64 x 32 f32 = 512 float4, 4 per thread. NT hint: stream-once data.
#pragma unroll
  for (int i = 0; i < 4; ++i) {
    int j   = tid + i * 128;
    int row = j >> 3;
    int c   = (j & 7) << 2;
    const float* g = A + (size_t)(mBase + row) * kN + k0 + c;
    __builtin_amdgcn_global_load_async_to_lds_b128(
        ASYNC_GPTR(g), ASYNC_LPTR(bA + row * kLda + c),
        0, /*cpol: TH_NT*/ 1);
  }
  // S slab: contiguous 32*128 f32 = 1024 float4, 8 per thread.
  const float* sgl = S + (size_t)k0 * kD;
#pragma unroll
  for (int i = 0; i < 8; ++i) {
    int j   = tid + i * 128;
    int row = j >> 5;
    int c   = (j & 31) << 2;
    __builtin_amdgcn_global_load_async_to_lds_b128(
        ASYNC_GPTR(sgl + ((size_t)j << 2)), ASYNC_LPTR(bS + row * kLds + c),
        0, 0);
  }
}

__global__ __launch_bounds__(128, 1) void gcn_aggregate(const float* __restrict__ A,
                                                        const float* __restrict__ S,
                                                        float* __restrict__ out) {
  __shared__ float sAb[2][kBlockM * kLda];   // 2 x 9216 B
  __shared__ float sSb[2][kKC * kLds];       // 2 x 17408 B

  const int tid  = threadIdx.x;
  const int wave = tid >> 5;
  const int lane = tid & 31;
  const int lm   = lane & 15;
  const int half = lane >> 4;
  const int mBase = blockIdx.x * kBlockM;

  v8f acc[8];
#pragma unroll
  for (int i = 0; i < 8; ++i) acc[i] = v8f{};

  // prologue: stage chunk 0 into buffer 0 (12 async instructions per thread)
  stage_async(A, S, mBase, 0, tid, sAb[0], sSb[0]);

  for (int c = 0; c < kChunks; ++c) {
    const int p = c & 1;
    if (c + 1 < kChunks) {
      // stage next chunk into the other buffer, then wait for current chunk:
      // async loads retire in order, so asynccnt<=12 means chunk c is done.
      stage_async(A, S, mBase, (c + 1) * kKC, tid, sAb[p ^ 1], sSb[p ^ 1]);
      __builtin_amdgcn_s_wait_asynccnt(12);
    } else {
      __builtin_amdgcn_s_wait_asynccnt(0);
    }
    __syncthreads();   // chunk c visible to all waves

    const float* ap  = &sAb[p][(wave * 16 + lm) * kLda + 2 * half];
    const float* bp0 = &sSb[p][(2 * half) * kLds + lm];
#pragma unroll
    for (int kk = 0; kk < kKC; kk += 4) {
      v2f a = *(const v2f*)(ap + kk);          // A[m][kk+2h], A[m][kk+2h+1]
#pragma unroll
      for (int nt = 0; nt < 8; ++nt) {
        v2f b;
        b.x = bp0[(size_t)kk * kLds + nt * 16];        // S[kk+2h][n]
        b.y = bp0[(size_t)(kk + 1) * kLds + nt * 16];  // S[kk+2h+1][n]
        acc[nt] = __builtin_amdgcn_wmma_f32_16x16x4_f32(false, a, false, b,
                                                        (short)0, acc[nt],
                                                        false, false);
      }
    }
    __syncthreads();   // buffer p free for reuse at iteration c+2
  }

  float* op = out + (size_t)(mBase + wave * 16 + 8 * half) * kD + lm;
#pragma unroll
  for (int nt = 0; nt < 8; ++nt) {
#pragma unroll
    for (int i = 0; i < 8; ++i) {
      op[(size_t)i * kD + nt * 16] = acc[nt][i];
    }
  }
}

#else  // !HAVE_ASYNC_LDS — synchronous single-buffer fallback (Round-1 body)

__global__ __launch_bounds__(128, 1) void gcn_aggregate(const float* __restrict__ A,
                                                        const float* __restrict__ S,
                                                        float* __restrict__ out) {
  __shared__ float sA[kBlockM * kLda];
  __shared__ float sS[kKC * kLds];

  const int tid  = threadIdx.x;
  const int wave = tid >> 5;
  const int lane = tid & 31;
  const int lm   = lane & 15;
  const int half = lane >> 4;
  const int mBase = blockIdx.x * kBlockM;

  v8f acc[8];
#pragma unroll
  for (int i = 0; i < 8; ++i) acc[i] = v8f{};

  for (int k0 = 0; k0 < kN; k0 += kKC) {
#pragma unroll
    for (int i = 0; i < 4; ++i) {
      int j   = tid + i * 128;
      int row = j >> 3;
      int c   = (j & 7) << 2;
      v4f v = __builtin_nontemporal_load(
          (const v4f*)(A + (size_t)(mBase + row) * kN + k0 + c));
      *(v4f*)&sA[row * kLda + c] = v;
    }
    const float* sgl = S + (size_t)k0 * kD;
#pragma unroll
    for (int i = 0; i < 8; ++i) {
      int j   = tid + i * 128;
      int row = j >> 5;
      int c   = (j & 31) << 2;
      v4f v = *(const v4f*)(sgl + ((size_t)j << 2));
      *(v4f*)&sS[row * kLds + c] = v;
    }
    __syncthreads();

    const float* ap  = &sA[(wave * 16 + lm) * kLda + 2 * half];
    const float* bp0 = &sS[(2 * half) * kLds + lm];
#pragma unroll
    for (int kk = 0; kk < kKC; kk += 4) {
      v2f a = *(const v2f*)(ap + kk);
#pragma unroll
      for (int nt = 0; nt < 8; ++nt) {
        v2f b;
        b.x = bp0[(size_t)kk * kLds + nt * 16];
        b.y = bp0[(size_t)(kk + 1) * kLds + nt * 16];
        acc[nt] = __builtin_amdgcn_wmma_f32_16x16x4_f32(false, a, false, b,
                                                        (short)0, acc[nt],
                                                        false, false);
      }
    }
    __syncthreads();
  }

  float* op = out + (size_t)(mBase + wave * 16 + 8 * half) * kD + lm;
#pragma unroll
  for (int nt = 0; nt < 8; ++nt) {
#pragma unroll
    for (int i = 0; i < 8; ++i) {
      op[(size_t)i * kD + nt * 16] = acc[nt][i];
    }
  }
}

#endif  // HAVE_ASYNC_LDS

// ---------------------------------------------------------------------------
extern "C" void kernel_launch(void* const* d_in, const int* in_sizes, int n_in,
                              void* d_out, int out_size, void* d_ws, size_t ws_size,
                              hipStream_t stream) {
  const float* A = (const float*)d_in[0];   // adjacency [16384,16384]
  const float* X = (const float*)d_in[1];   // input_feature [16384,128]
  const float* W = (const float*)d_in[2];   // weight [128,128]
  float* out = (float*)d_out;               // [16384,128]
  float* S   = (float*)d_ws;                // support scratch: 8 MB

  dim3 b1(128);
  dim3 g1((kN / 16) * (kD / 16) / 4);       // 2048 blocks, 1 wave per tile
  hipLaunchKernelGGL(gcn_support, g1, b1, 0, stream, X, W, S);

  dim3 b2(128);
  dim3 g2(kN / kBlockM);                    // 256 blocks
  hipLaunchKernelGGL(gcn_aggregate, g2, b2, 0, stream, A, S, out);
}